// CLS_57664230916483
// MI455X (gfx1250) — compile-verified
//
#include <hip/hip_runtime.h>

#define N_NODES 100000
#define N_EDGES 3200000
#define IN_DIM  256
#define OUT_DIM 64

typedef __attribute__((ext_vector_type(2))) float v2f;
typedef __attribute__((ext_vector_type(8))) float v8f;

// ---------------- utility ----------------
__global__ void zero_f32(float* __restrict__ p, int n) {
    int i = blockIdx.x * blockDim.x + threadIdx.x;
    if (i < n) p[i] = 0.0f;
}

// ---------------- degrees ----------------
__global__ void degree_kernel(const long long* __restrict__ edges,
                              float* __restrict__ deg_src,
                              float* __restrict__ deg_dst) {
    int e = blockIdx.x * blockDim.x + threadIdx.x;
    if (e < N_EDGES) {
        int s = (int)edges[e];
        int d = (int)edges[(size_t)N_EDGES + e];
        atomicAdd(&deg_src[s], 1.0f);
        atomicAdd(&deg_dst[d], 1.0f);
    }
}

__global__ void rsqrt_clamp_kernel(float* __restrict__ p, int n) {
    int i = blockIdx.x * blockDim.x + threadIdx.x;
    if (i < n) {
        float v = fmaxf(p[i], 1.0f);
        p[i] = 1.0f / sqrtf(v);
    }
}

// ---------------- GEMM: m = (h @ W) * norm_src, via V_WMMA_F32_16X16X4_F32 ----------------
// One wave computes a 16x64 output tile (four 16x16 WMMA accumulators).
// A layout (16x4 f32): lane L<16 -> row L, K pair (k,k+1); lane>=16 -> row L-16, K pair (k+2,k+3).
// B layout (4x16 f32): vgpr0 = row K(+2 for hi half), vgpr1 = row K+1(+2), N = lane&15.
// W is staged in LDS K-pair interleaved: sW2[p*64+n] = {W[2p][n], W[2p+1][n]},
// so every B operand is ONE aligned ds_load_b64 straight into a WMMA-ready pair.
__global__ __launch_bounds__(256) void gemm_wmma_f32(
        const float* __restrict__ h,
        const float* __restrict__ W,
        const float* __restrict__ norm_src,
        float* __restrict__ m) {
    __shared__ v2f sW2[(IN_DIM / 2) * OUT_DIM];   // 64 KB of the 320 KB WGP LDS

    // cooperative interleaved stage of W
    for (int i = threadIdx.x; i < (IN_DIM / 2) * OUT_DIM; i += blockDim.x) {
        int p = i >> 6;          // k-pair index 0..127
        int n = i & 63;          // column
        v2f v;
        v.x = W[(2 * p)     * OUT_DIM + n];
        v.y = W[(2 * p + 1) * OUT_DIM + n];
        sW2[i] = v;
    }
    __syncthreads();

    const int wave = threadIdx.x >> 5;
    const int lane = threadIdx.x & 31;
    const int tile = blockIdx.x * 8 + wave;
    if (tile >= N_NODES / 16) return;

    const int row0 = tile * 16;
    const int hi   = lane >> 4;      // 0: lower half-wave, 1: upper
    const int lo   = lane & 15;
    const int koff = hi ? 2 : 0;

    v8f acc0 = {}, acc1 = {}, acc2 = {}, acc3 = {};
    const float* hrow = h + (size_t)(row0 + lo) * IN_DIM + koff;

    #pragma unroll 4
    for (int k = 0; k < IN_DIM; k += 4) {
        v2f a;
        a.x = hrow[k];
        a.y = hrow[k + 1];

        const v2f* wrow = &sW2[((k >> 1) + hi) * OUT_DIM + lo];
        v2f b0 = wrow[0];
        v2f b1 = wrow[16];
        v2f b2 = wrow[32];
        v2f b3 = wrow[48];

        acc0 = __builtin_amdgcn_wmma_f32_16x16x4_f32(false, a, false, b0, (short)0, acc0, false, false);
        acc1 = __builtin_amdgcn_wmma_f32_16x16x4_f32(false, a, false, b1, (short)0, acc1, false, false);
        acc2 = __builtin_amdgcn_wmma_f32_16x16x4_f32(false, a, false, b2, (short)0, acc2, false, false);
        acc3 = __builtin_amdgcn_wmma_f32_16x16x4_f32(false, a, false, b3, (short)0, acc3, false, false);
    }

    // scale rows by norm_src and store to m
    float* mout = m + (size_t)row0 * OUT_DIM;
    #pragma unroll
    for (int j = 0; j < 8; ++j) {
        int r = j + 8 * hi;
        float nv = norm_src[row0 + r];
        float* rowp = mout + (size_t)r * OUT_DIM + lo;
        rowp[0]  = acc0[j] * nv;
        rowp[16] = acc1[j] * nv;
        rowp[32] = acc2[j] * nv;
        rowp[48] = acc3[j] * nv;
    }
}

// ---------------- SpMM scatter: agg[dst] += m[src] ----------------
// 16 threads per edge, 4 floats each (float4 gather, 4 global_atomic_add_f32).
// m (25.6 MB) and agg (25.6 MB) both fit in the 192 MB L2.
__global__ void spmm_scatter(const long long* __restrict__ edges,
                             const float* __restrict__ m,
                             float* __restrict__ agg) {
    int tid = blockIdx.x * blockDim.x + threadIdx.x;
    int e  = tid >> 4;
    int c4 = (tid & 15) << 2;
    if (e < N_EDGES) {
        int s = (int)edges[e];
        int d = (int)edges[(size_t)N_EDGES + e];
        const float4 v = *(const float4*)&m[(size_t)s * OUT_DIM + c4];
        float* dst = &agg[(size_t)d * OUT_DIM + c4];
        atomicAdd(dst + 0, v.x);
        atomicAdd(dst + 1, v.y);
        atomicAdd(dst + 2, v.z);
        atomicAdd(dst + 3, v.w);
    }
}

// ---------------- finalize: out = log_softmax(agg * norm_dst + b) ----------------
// One wave32 per row; each lane owns cols (lane, lane+32); shfl_xor reductions.
__global__ __launch_bounds__(256) void finalize_kernel(
        float* __restrict__ out,
        const float* __restrict__ norm_dst,
        const float* __restrict__ bias) {
    const int wave = threadIdx.x >> 5;
    const int lane = threadIdx.x & 31;
    const int row  = blockIdx.x * 8 + wave;
    if (row >= N_NODES) return;

    const float nd = norm_dst[row];
    float* p = out + (size_t)row * OUT_DIM;
    float v0 = p[lane]      * nd + bias[lane];
    float v1 = p[lane + 32] * nd + bias[lane + 32];

    float mx = fmaxf(v0, v1);
    #pragma unroll
    for (int off = 16; off > 0; off >>= 1)
        mx = fmaxf(mx, __shfl_xor(mx, off, 32));

    float s = expf(v0 - mx) + expf(v1 - mx);
    #pragma unroll
    for (int off = 16; off > 0; off >>= 1)
        s += __shfl_xor(s, off, 32);

    const float lse = mx + logf(s);
    p[lane]      = v0 - lse;
    p[lane + 32] = v1 - lse;
}

// ---------------- launch ----------------
extern "C" void kernel_launch(void* const* d_in, const int* in_sizes, int n_in,
                              void* d_out, int out_size, void* d_ws, size_t ws_size,
                              hipStream_t stream) {
    const float*     h     = (const float*)d_in[0];
    const float*     W     = (const float*)d_in[1];
    const float*     b     = (const float*)d_in[2];
    const long long* edges = (const long long*)d_in[3];
    float* out = (float*)d_out;

    float* ws       = (float*)d_ws;
    float* norm_src = ws;                    // [N_NODES]
    float* norm_dst = ws + N_NODES;          // [N_NODES]
    float* m        = ws + 2 * N_NODES;      // [N_NODES * OUT_DIM]

    // zero degree accumulators and output accumulator
    zero_f32<<<(2 * N_NODES + 255) / 256, 256, 0, stream>>>(norm_src, 2 * N_NODES);
    zero_f32<<<(N_NODES * OUT_DIM + 255) / 256, 256, 0, stream>>>(out, N_NODES * OUT_DIM);

    // degrees -> clamped rsqrt norms
    degree_kernel<<<(N_EDGES + 255) / 256, 256, 0, stream>>>(edges, norm_src, norm_dst);
    rsqrt_clamp_kernel<<<(2 * N_NODES + 255) / 256, 256, 0, stream>>>(ws, 2 * N_NODES);

    // m = (h @ W) * norm_src   (fp32 WMMA; 8 waves/block, 1 tile of 16 rows per wave)
    gemm_wmma_f32<<<(N_NODES / 16 + 7) / 8, 256, 0, stream>>>(h, W, norm_src, m);

    // agg[dst] += m[src]  (into d_out)
    spmm_scatter<<<(N_EDGES * 16) / 256, 256, 0, stream>>>(edges, m, out);

    // out = log_softmax(agg * norm_dst + b)
    finalize_kernel<<<N_NODES / 8, 256, 0, stream>>>(out, norm_dst, b);
}